// SmoothnessLoss_31129922961650
// MI455X (gfx1250) — compile-verified
//
#include <hip/hip_runtime.h>
#include <stdint.h>
#include <math.h>

// ---------------------------------------------------------------------------
// SmoothnessLoss for 1x3x224x224 f32 image pair, 21x21 window.
// Each 256-thread block computes a 16x16 pixel tile. Halos of both images are
// staged in LDS as float4 (rgb0) via CDNA5 async global->LDS loads, so the
// hot loops use ds_load_b128 (one per image per neighbor) and native
// v_exp/v_log transcendentals.
// ---------------------------------------------------------------------------

namespace {

constexpr int WSR   = 10;            // window radius
constexpr int TX    = 16;
constexpr int TY    = 16;
constexpr int HX    = TX + 2 * WSR;  // 36
constexpr int HY    = TY + 2 * WSR;  // 36
constexpr int IMGW  = 224;
constexpr int IMGH  = 224;
constexpr int PLANE = IMGW * IMGH;
constexpr int NBLKX = IMGW / TX;     // 14
constexpr int NBLKY = IMGH / TY;     // 14
constexpr int NBLK  = NBLKX * NBLKY; // 196

constexpr float SIG_COLOR = 50.0f;          // 1/(0.1^2 * 2)
constexpr float SIG_SPACE = 1.0f / 98.0f;   // 1/(7^2 * 2)
constexpr float LPW       = 0.8f;
constexpr float C1V       = 10.0f;
constexpr float C2V       = 5.0f;
constexpr float ALPHAV    = 5.0f;
constexpr float EPSV      = 1e-6f;
// scale = 1/(h*w*b)/(WS*WS)  (reference quirk: WS^2, not K^2)
constexpr float SCALEV    = 1.0f / (224.0f * 224.0f * 100.0f);

// Low 32 bits of a flat pointer to __shared__ == LDS byte address.
__device__ __forceinline__ uint32_t lds_addr32(const void* p) {
  return (uint32_t)(uintptr_t)p;
}

// CDNA5 async DMA: global -> LDS, 4 bytes per active lane (ASYNCcnt).
__device__ __forceinline__ void async_gl2lds_b32(uint32_t lds_addr,
                                                 const float* sbase,
                                                 uint32_t voff_bytes) {
  asm volatile("global_load_async_to_lds_b32 %0, %1, %2"
               :: "v"(lds_addr), "v"(voff_bytes), "s"(sbase)
               : "memory");
}

__device__ __forceinline__ void wait_async0() {
  asm volatile("s_wait_asynccnt 0" ::: "memory");
}

// t^0.8 for t in [1e-6, ~1]: native exp2/log2 path (no edge cases possible).
__device__ __forceinline__ float fast_pow08(float t) {
  return __expf(LPW * __logf(t));
}

__global__ __launch_bounds__(TX * TY) void smooth_loss_main(
    const float* __restrict__ orig,
    const float* __restrict__ smth,
    float* __restrict__ partials) {

  __shared__ float4 sO4[HY][HX];    // original halo tile, rgb0 per texel
  __shared__ float4 sS4[HY][HX];    // smooth halo tile, rgb0 per texel
  __shared__ float  e1[WSR + 1];    // 1-D spatial gaussian
  __shared__ float  redw[8];

  const int tx  = threadIdx.x;
  const int ty  = threadIdx.y;
  const int tid = ty * TX + tx;
  const int gx0 = (int)blockIdx.x * TX - WSR;
  const int gy0 = (int)blockIdx.y * TY - WSR;

  if (tid <= WSR) {
    float d = (float)tid;
    e1[tid] = __expf(-SIG_SPACE * d * d);
  }

  // ---- stage halos: async-load in-bounds cells, ds-store zeros otherwise --
  // slot index: {img, y, x, c} with c in 0..3 (c==3 is the zero pad lane)
  constexpr int SLOTS = 2 * HY * HX * 4;   // 10368
  for (int i = tid; i < SLOTS; i += TX * TY) {
    int img = i / (HY * HX * 4);
    int rem = i - img * (HY * HX * 4);
    int c   = rem & 3;
    int p   = rem >> 2;
    int ly  = p / HX;
    int lx  = p - ly * HX;
    int gx  = gx0 + lx;
    int gy  = gy0 + ly;
    float* dst = (float*)(img ? &sS4[ly][lx] : &sO4[ly][lx]) + c;
    if (c < 3 && (unsigned)gx < (unsigned)IMGW && (unsigned)gy < (unsigned)IMGH) {
      const float* src = img ? smth : orig;
      uint32_t off = (uint32_t)((c * PLANE + gy * IMGW + gx) * 4);
      async_gl2lds_b32(lds_addr32(dst), src, off);
    } else {
      *dst = 0.0f;   // zero padding (disjoint addresses from async targets)
    }
  }
  wait_async0();     // this wave's async writes landed in LDS
  __syncthreads();   // + DS waits; all waves' tiles visible

  const int lx = tx + WSR;
  const int ly = ty + WSR;
  const float4 oc = sO4[ly][lx];
  const float4 sc = sS4[ly][lx];

  // ---- pass 1: edge responses (masked |diff| sums over window+channels) ----
  float er_o = 0.0f, er_s = 0.0f;
  for (int dy = -WSR; dy <= WSR; ++dy) {
    const int yy = ly + dy;
    for (int dx = -WSR; dx <= WSR; ++dx) {
      const int xx = lx + dx;
      float4 o = sO4[yy][xx];
      float4 s = sS4[yy][xx];
      er_o += (o.x > 0.0f ? fabsf(o.x - oc.x) : 0.0f)
            + (o.y > 0.0f ? fabsf(o.y - oc.y) : 0.0f)
            + (o.z > 0.0f ? fabsf(o.z - oc.z) : 0.0f);
      er_s += (s.x > 0.0f ? fabsf(s.x - sc.x) : 0.0f)
            + (s.y > 0.0f ? fabsf(s.y - sc.y) : 0.0f)
            + (s.z > 0.0f ? fabsf(s.z - sc.z) : 0.0f);
    }
  }
  const bool use_large = (er_o < C1V) && ((er_s - er_o) > C2V);

  // ---- pass 2: loss terms (branch hoisted out of the 441-iter loop) ----
  float acc = 0.0f;
  if (use_large) {
    for (int dy = -WSR; dy <= WSR; ++dy) {
      const int yy = ly + dy;
      const float ey = ALPHAV * e1[dy < 0 ? -dy : dy];
      for (int dx = -WSR; dx <= WSR; ++dx) {
        const int xx = lx + dx;
        const float w = ey * e1[dx < 0 ? -dx : dx];
        float4 s = sS4[yy][xx];
        float t0 = (s.x > 0.0f) ? fabsf(s.x - sc.x) : 0.0f;
        float t1 = (s.y > 0.0f) ? fabsf(s.y - sc.y) : 0.0f;
        float t2 = (s.z > 0.0f) ? fabsf(s.z - sc.z) : 0.0f;
        acc += w * (t0 * t0 + t1 * t1 + t2 * t2);
      }
    }
  } else {
    for (int dy = -WSR; dy <= WSR; ++dy) {
      const int yy = ly + dy;
      for (int dx = -WSR; dx <= WSR; ++dx) {
        const int xx = lx + dx;
        float4 o = sO4[yy][xx];
        // color distance is UNMASKED in the reference (OOB zeros included)
        float d0 = o.x - oc.x, d1 = o.y - oc.y, d2 = o.z - oc.z;
        float wr = __expf(-SIG_COLOR * (d0 * d0 + d1 * d1 + d2 * d2));
        float4 s = sS4[yy][xx];
        float t0 = (s.x > 0.0f) ? fabsf(s.x - sc.x) : 0.0f;
        float t1 = (s.y > 0.0f) ? fabsf(s.y - sc.y) : 0.0f;
        float t2 = (s.z > 0.0f) ? fabsf(s.z - sc.z) : 0.0f;
        float tp = fast_pow08(t0 + EPSV) + fast_pow08(t1 + EPSV)
                 + fast_pow08(t2 + EPSV);
        acc += wr * tp;
      }
    }
  }

  // ---- block reduction: wave32 shuffle, then cross-wave via LDS ----
  for (int off = 16; off > 0; off >>= 1) acc += __shfl_down(acc, off);
  if ((tid & 31) == 0) redw[tid >> 5] = acc;
  __syncthreads();
  if (tid == 0) {
    float s = 0.0f;
#pragma unroll
    for (int i = 0; i < 8; ++i) s += redw[i];
    partials[blockIdx.y * gridDim.x + blockIdx.x] = s;
  }
}

__global__ __launch_bounds__(256) void smooth_loss_reduce(
    const float* __restrict__ partials, int n, float* __restrict__ out) {
  __shared__ float red[256];
  float s = 0.0f;
  for (int i = threadIdx.x; i < n; i += 256) s += partials[i];
  red[threadIdx.x] = s;
  __syncthreads();
  for (int k = 128; k > 0; k >>= 1) {
    if ((int)threadIdx.x < k) red[threadIdx.x] += red[threadIdx.x + k];
    __syncthreads();
  }
  if (threadIdx.x == 0) out[0] = red[0] * SCALEV;
}

}  // namespace

extern "C" void kernel_launch(void* const* d_in, const int* in_sizes, int n_in,
                              void* d_out, int out_size, void* d_ws, size_t ws_size,
                              hipStream_t stream) {
  (void)in_sizes; (void)n_in; (void)out_size; (void)ws_size;
  const float* orig = (const float*)d_in[0];   // original_images
  const float* smth = (const float*)d_in[1];   // smooth_images
  float* partials = (float*)d_ws;              // 196 floats of scratch
  dim3 grid(NBLKX, NBLKY), block(TX, TY);
  smooth_loss_main<<<grid, block, 0, stream>>>(orig, smth, partials);
  smooth_loss_reduce<<<1, 256, 0, stream>>>(partials, NBLK, (float*)d_out);
}